// EfficientDetWithPostProcess_79448305042063
// MI455X (gfx1250) — compile-verified
//
#include <hip/hip_runtime.h>
#include <hip/hip_bf16.h>
#include <stdint.h>

#define N_CLASSES 90
#define TILE 128          // anchors per block in decode stage
#define NMS_BLOCKS 240
#define NMS_THREADS 256
#define MAX_DET 100
#define NEG_SCORE (-1e9f)

typedef unsigned int u32x4 __attribute__((ext_vector_type(4)));
typedef int          i32x8 __attribute__((ext_vector_type(8)));
typedef int          i32x4 __attribute__((ext_vector_type(4)));

// ---------- helpers ----------
__device__ __forceinline__ unsigned enc_score(float f) {
    unsigned u = __float_as_uint(f);
    return (u & 0x80000000u) ? ~u : (u | 0x80000000u);
}
__device__ __forceinline__ float dec_score(unsigned u) {
    unsigned b = (u & 0x80000000u) ? (u & 0x7fffffffu) : ~u;
    return __uint_as_float(b);
}

// ---------- init (graph-replay safe re-initialization of winner slots) ----------
__global__ void init_best_kernel(unsigned long long* best) {
    if (threadIdx.x < MAX_DET) best[threadIdx.x] = 0ull;
}

// ---------- stage 1: decode boxes + per-anchor class max/argmax ----------
// The classification tile (rows x 90 f32) is DMA'd into LDS by the CDNA5
// Tensor Data Mover: one tensor_load_to_lds per block, issued by wave 0,
// completion tracked with TENSORcnt. The D# descriptor describes a 2-D
// tensor of width 90 elements (data_size = 4B), stride 90, tile = rows x 90,
// landing contiguously in LDS (row-major, 360B rows).
__global__ __launch_bounds__(TILE) void decode_kernel(
    const float* __restrict__ reg, const float* __restrict__ cls,
    const float* __restrict__ anc,
    const int* __restrict__ img_h, const int* __restrict__ img_w,
    float* __restrict__ boxes,
    float* __restrict__ smask, int* __restrict__ classes, int N)
{
    __shared__ float sh[TILE * N_CLASSES];   // 46080 bytes

    const int tid   = threadIdx.x;
    const int tile0 = blockIdx.x * TILE;
    const int rows  = min(TILE, N - tile0);

    if (tid < 32) {  // wave 0 issues the TDM transfer for the block
        const unsigned long long ga =
            (unsigned long long)(uintptr_t)cls + (unsigned long long)tile0 * 360ull;
        const unsigned lds = (unsigned)(uintptr_t)(&sh[0]);

        u32x4 g0;
        g0[0] = 1u;                                    // count=1 (valid), user mode
        g0[1] = lds;                                   // lds_addr [63:32]
        g0[2] = (unsigned)(ga & 0xffffffffull);        // global_addr[31:0]
        g0[3] = ((unsigned)(ga >> 32) & 0x01ffffffu)   // global_addr[56:32]
                | 0x80000000u;                         // type=2 ("image")

        i32x8 g1;
        g1[0] = 0x00020000;                            // wg_mask=0, data_size=2 (4B)
        g1[1] = (int)(90u << 16);                      // tensor_dim0 = 90  [63:48]
        g1[2] = (int)((unsigned)rows << 16);           // tensor_dim1[15:0] [95:80]
        g1[3] = (int)(90u << 16);                      // tile_dim0 = 90    [127:112]
        g1[4] = rows;                                  // tile_dim1 = rows; tile_dim2=0
        g1[5] = 90;                                    // tensor_dim0_stride = 90
        g1[6] = 0;
        g1[7] = 0;

        i32x4 g2 = {0, 0, 0, 0};                       // 2-D tensor: dims 2/3 unused
        i32x4 g3 = {0, 0, 0, 0};
        i32x8 g4 = {0, 0, 0, 0, 0, 0, 0, 0};           // unused slot (non-gather mode)

        __builtin_amdgcn_tensor_load_to_lds(g0, g1, g2, g3, g4, 0);
        __builtin_amdgcn_s_wait_tensorcnt(0);
    }
    __syncthreads();

    if (tid < rows) {
        const int i = tile0 + tid;
        // max + first-occurrence argmax over 90 classes (float2 LDS reads:
        // 8B-aligned rows, stride-90-dword rows are bank-conflict free)
        const float2* srow = (const float2*)(&sh[tid * N_CLASSES]);
        float2 v0 = srow[0];
        float mx = v0.x;
        int   mi = 0;
        if (v0.y > mx) { mx = v0.y; mi = 1; }
        #pragma unroll 4
        for (int k = 1; k < N_CLASSES / 2; ++k) {
            float2 v = srow[k];
            if (v.x > mx) { mx = v.x; mi = 2 * k; }
            if (v.y > mx) { mx = v.y; mi = 2 * k + 1; }
        }

        // decode + clip (BBoxTransform std = 0.1/0.1/0.2/0.2, mean 0)
        const float4 a  = ((const float4*)anc)[i];
        const float4 rg = ((const float4*)reg)[i];
        const float W = (float)img_w[0];
        const float H = (float)img_h[0];
        const float wa = a.z - a.x;
        const float ha = a.w - a.y;
        const float cxa = a.x + 0.5f * wa;
        const float cya = a.y + 0.5f * ha;
        const float cx = cxa + rg.x * 0.1f * wa;
        const float cy = cya + rg.y * 0.1f * ha;
        const float w  = expf(rg.z * 0.2f) * wa;
        const float h  = expf(rg.w * 0.2f) * ha;
        const float x1 = fmaxf(cx - 0.5f * w, 0.0f);
        const float y1 = fmaxf(cy - 0.5f * h, 0.0f);
        const float x2 = fminf(cx + 0.5f * w, W);
        const float y2 = fminf(cy + 0.5f * h, H);

        ((float4*)boxes)[i] = make_float4(x1, y1, x2, y2);
        smask[i]   = (mx > 0.05f) ? mx : NEG_SCORE;
        classes[i] = mi;
    }
}

// ---------- stage 2: one fused NMS round ----------
// Round r: suppress every box overlapping winner of round r-1 (iou > 0.5,
// applied unconditionally as in the reference; self IoU==1 retires the
// winner), while simultaneously computing the device-wide argmax of the
// updated masked scores into best[r] via packed u64 atomicMax (float score
// order-encoded in the high word, ~index in the low word so ties pick the
// lowest index, matching jnp.argmax).
__global__ __launch_bounds__(NMS_THREADS) void nms_round_kernel(
    float* __restrict__ smask, const float4* __restrict__ boxes4,
    unsigned long long* __restrict__ best, int r, int N)
{
    __shared__ unsigned long long red[NMS_THREADS];

    float4 pb = make_float4(0.f, 0.f, 0.f, 0.f);
    float parea = 0.f;
    const bool havePrev = (r > 0);
    if (havePrev) {
        const unsigned long long pk = best[r - 1];
        const int pidx = (int)(~(unsigned)pk);
        pb    = boxes4[pidx];
        parea = (pb.z - pb.x) * (pb.w - pb.y);
    }

    unsigned long long lk = 0ull;
    const int stride = gridDim.x * blockDim.x;
    for (int i = blockIdx.x * blockDim.x + threadIdx.x; i < N; i += stride) {
        float s = smask[i];
        if (havePrev) {
            const float4 b = boxes4[i];
            const float ai  = (b.z - b.x) * (b.w - b.y);
            const float xx1 = fmaxf(pb.x, b.x);
            const float yy1 = fmaxf(pb.y, b.y);
            const float xx2 = fminf(pb.z, b.z);
            const float yy2 = fminf(pb.w, b.w);
            const float inter = fmaxf(xx2 - xx1, 0.0f) * fmaxf(yy2 - yy1, 0.0f);
            const float iou = inter / (((parea + ai) - inter) + 1e-8f);
            if (iou > 0.5f) { s = NEG_SCORE; smask[i] = s; }
        }
        const unsigned long long key =
            ((unsigned long long)enc_score(s) << 32) |
            (unsigned long long)(~(unsigned)i);
        lk = (key > lk) ? key : lk;
    }

    red[threadIdx.x] = lk;
    __syncthreads();
    #pragma unroll
    for (int o = NMS_THREADS / 2; o > 0; o >>= 1) {
        if (threadIdx.x < o) {
            unsigned long long other = red[threadIdx.x + o];
            if (other > red[threadIdx.x]) red[threadIdx.x] = other;
        }
        __syncthreads();
    }
    if (threadIdx.x == 0) atomicMax(best + r, red[0]);
}

// ---------- stage 3: gather outputs ----------
// d_out layout (flat, reference return order, all as float32):
//   [0,400)   out_boxes  [100,4]
//   [400,500) out_scores [100]
//   [500,600) out_classes[100]
__global__ void gather_kernel(const unsigned long long* __restrict__ best,
                              const float4* __restrict__ boxes4,
                              const int* __restrict__ classes,
                              float* __restrict__ out)
{
    const int k = threadIdx.x;
    if (k >= MAX_DET) return;
    const unsigned long long key = best[k];
    const unsigned hi = (unsigned)(key >> 32);
    const unsigned lo = (unsigned)key;
    const float s = dec_score(hi);
    const bool valid = (s > NEG_SCORE * 0.5f);   // s > -5e8
    float4 b = make_float4(0.f, 0.f, 0.f, 0.f);
    float sc = 0.f, cl = 0.f;
    if (valid) {
        const int idx = (int)(~lo);
        b  = boxes4[idx];
        sc = s;
        cl = (float)classes[idx];
    }
    out[k * 4 + 0] = b.x;
    out[k * 4 + 1] = b.y;
    out[k * 4 + 2] = b.z;
    out[k * 4 + 3] = b.w;
    out[400 + k] = sc;
    out[500 + k] = cl;
}

extern "C" void kernel_launch(void* const* d_in, const int* in_sizes, int n_in,
                              void* d_out, int out_size, void* d_ws, size_t ws_size,
                              hipStream_t stream) {
    (void)n_in; (void)out_size; (void)ws_size;

    const float* reg = (const float*)d_in[0];   // [1,N,4]
    const float* cls = (const float*)d_in[1];   // [1,N,90]
    const float* anc = (const float*)d_in[2];   // [1,N,4]
    const int*   ih  = (const int*)d_in[3];     // scalar
    const int*   iw  = (const int*)d_in[4];     // scalar
    float* out = (float*)d_out;

    const int N = in_sizes[0] / 4;              // 196416

    // workspace layout (8B-aligned: N*24 bytes before `best`)
    float* boxes   = (float*)d_ws;              // N*4 floats
    float* smask   = boxes + (size_t)N * 4;     // N floats
    int*   classes = (int*)(smask + N);         // N ints
    unsigned long long* best = (unsigned long long*)(classes + N); // 100 u64

    init_best_kernel<<<1, 128, 0, stream>>>(best);

    const int dblocks = (N + TILE - 1) / TILE;
    decode_kernel<<<dblocks, TILE, 0, stream>>>(reg, cls, anc, ih, iw,
                                                boxes, smask, classes, N);

    for (int r = 0; r < MAX_DET; ++r) {
        nms_round_kernel<<<NMS_BLOCKS, NMS_THREADS, 0, stream>>>(
            smask, (const float4*)boxes, best, r, N);
    }

    gather_kernel<<<1, 128, 0, stream>>>(best, (const float4*)boxes, classes, out);
}